// IMUKalmanFilter_50345606643833
// MI455X (gfx1250) — compile-verified
//
#include <hip/hip_runtime.h>
#include <math.h>

#if defined(__HIP_DEVICE_COMPILE__)
#if !__has_builtin(__builtin_amdgcn_wmma_f32_16x16x4_f32)
#error "missing __builtin_amdgcn_wmma_f32_16x16x4_f32 on this toolchain (device)"
#endif
#endif

typedef __attribute__((ext_vector_type(2))) float v2f;
typedef __attribute__((ext_vector_type(8))) float v8f;

#define MS 34                 // padded row stride (floats) for 32x32 LDS tiles
#define BUFSZ (32 * MS)       // one padded 32x32 matrix
#define WPB 2                 // waves per block (2 * ~24KB LDS < 64KB static)

// ---------------- small 3x3 helpers (per-lane redundant, wave-uniform) ----------
__device__ __forceinline__ void m3mul(const float* A, const float* B, float* C) {
#pragma unroll
  for (int i = 0; i < 3; i++)
#pragma unroll
    for (int j = 0; j < 3; j++)
      C[i*3+j] = A[i*3+0]*B[0*3+j] + A[i*3+1]*B[1*3+j] + A[i*3+2]*B[2*3+j];
}
__device__ __forceinline__ void m3mulBT(const float* A, const float* B, float* C) { // A @ B^T
#pragma unroll
  for (int i = 0; i < 3; i++)
#pragma unroll
    for (int j = 0; j < 3; j++)
      C[i*3+j] = A[i*3+0]*B[j*3+0] + A[i*3+1]*B[j*3+1] + A[i*3+2]*B[j*3+2];
}
__device__ __forceinline__ void m3vec(const float* A, const float* x, float* y) {
#pragma unroll
  for (int i = 0; i < 3; i++) y[i] = A[i*3+0]*x[0] + A[i*3+1]*x[1] + A[i*3+2]*x[2];
}
__device__ __forceinline__ void m3tvec(const float* A, const float* x, float* y) { // A^T x
#pragma unroll
  for (int i = 0; i < 3; i++) y[i] = A[0*3+i]*x[0] + A[1*3+i]*x[1] + A[2*3+i]*x[2];
}
__device__ __forceinline__ void m3T(const float* A, float* B) {
#pragma unroll
  for (int i = 0; i < 3; i++)
#pragma unroll
    for (int j = 0; j < 3; j++) B[i*3+j] = A[j*3+i];
}
__device__ __forceinline__ void skew3(const float* v, float* K) {
  K[0] = 0.f;    K[1] = -v[2];  K[2] = v[1];
  K[3] = v[2];   K[4] = 0.f;    K[5] = -v[0];
  K[6] = -v[1];  K[7] = v[0];   K[8] = 0.f;
}
__device__ __forceinline__ void expso3(const float* phi, float* R) {
  float th2 = phi[0]*phi[0] + phi[1]*phi[1] + phi[2]*phi[2];
  float th  = sqrtf(th2 + 1e-20f);
  bool  sm  = th2 < 1e-12f;
  float ths = sm ? 1.f : th;
  float A_  = sm ? (1.f - th2 / 6.f)  : (sinf(ths) / ths);
  float B_  = sm ? (0.5f - th2/24.f)  : ((1.f - cosf(ths)) / (ths*ths));
  float K[9], K2[9];
  skew3(phi, K);  m3mul(K, K, K2);
#pragma unroll
  for (int i = 0; i < 3; i++)
#pragma unroll
    for (int j = 0; j < 3; j++)
      R[i*3+j] = (i==j ? 1.f : 0.f) + A_*K[i*3+j] + B_*K2[i*3+j];
}
__device__ __forceinline__ void logso3(const float* R, float* phi) {
  float tr = R[0] + R[4] + R[8];
  float c  = fminf(fmaxf((tr - 1.f)*0.5f, -1.f + 1e-7f), 1.f - 1e-7f);
  float ph = acosf(c);
  bool  sm = ph < 1e-6f;
  float phs = sm ? 1.f : ph;
  float coef = sm ? (0.5f + ph*ph/12.f) : (phs / (2.f * sinf(phs)));
  phi[0] = coef * (R[7] - R[5]);
  phi[1] = coef * (R[2] - R[6]);
  phi[2] = coef * (R[3] - R[1]);
}
__device__ __forceinline__ void jleftinv(const float* phi, float* J) {
  float th2 = phi[0]*phi[0] + phi[1]*phi[1] + phi[2]*phi[2];
  float th  = sqrtf(th2 + 1e-20f);
  bool  sm  = th2 < 1e-12f;
  float ths = sm ? 1.f : th;
  float coef = sm ? (1.f/12.f)
                  : (1.f/(ths*ths) - (1.f + cosf(ths)) / (2.f*ths*sinf(ths)));
  float K[9], K2[9];
  skew3(phi, K);  m3mul(K, K, K2);
#pragma unroll
  for (int i = 0; i < 3; i++)
#pragma unroll
    for (int j = 0; j < 3; j++)
      J[i*3+j] = (i==j ? 1.f : 0.f) - 0.5f*K[i*3+j] + coef*K2[i*3+j];
}

// ---------------- LDS matrix helpers ----------------
__device__ __forceinline__ void zero32(float* buf, int lane) {
  for (int i = lane; i < BUFSZ; i += 32) buf[i] = 0.f;
}
__device__ __forceinline__ void putB3(float* M, int r0, int c0, const float* A, float s) {
#pragma unroll
  for (int r = 0; r < 3; r++)
#pragma unroll
    for (int c = 0; c < 3; c++) M[(r0+r)*MS + c0 + c] = s * A[r*3+c];
}
__device__ __forceinline__ void putI3(float* M, int r0, int c0, float s) {
#pragma unroll
  for (int r = 0; r < 3; r++) M[(r0+r)*MS + c0 + r] = s;
}

// D(32x32) = A(32x32) @ B(32x32) (optionally B^T), zero-padded operands, via
// V_WMMA_F32_16X16X4_F32: 2x2 result tiles, K marched in chunks of 4.
// A 16x4 lane layout: lanes 0-15 hold K=0,1 in v0,v1; lanes 16-31 hold K=2,3.
// B 4x16 / C-D layouts per CDNA5 ISA 7.12.2. D must not alias A or B.
__device__ inline void mm32(const float* A, const float* B, float* D,
                            bool transB, int lane) {
  const int half = lane >> 4;   // 0: K+0/K+1 ; 1: K+2/K+3
  const int l    = lane & 15;
#pragma unroll
  for (int ti = 0; ti < 2; ti++) {
#pragma unroll
    for (int tj = 0; tj < 2; tj++) {
      v8f c = {};
#pragma unroll
      for (int k = 0; k < 8; k++) {
        const int kb = k*4 + half*2;
        v2f a, b;
        a.x = A[(ti*16 + l)*MS + kb + 0];
        a.y = A[(ti*16 + l)*MS + kb + 1];
        if (transB) {
          b.x = B[(tj*16 + l)*MS + kb + 0];
          b.y = B[(tj*16 + l)*MS + kb + 1];
        } else {
          b.x = B[(kb + 0)*MS + tj*16 + l];
          b.y = B[(kb + 1)*MS + tj*16 + l];
        }
        c = __builtin_amdgcn_wmma_f32_16x16x4_f32(false, a, false, b,
                                                  (short)0, c, false, false);
      }
#pragma unroll
      for (int r = 0; r < 8; r++)   // VGPR r -> row r (lanes<16) / r+8 (lanes>=16)
        D[(ti*16 + r + 8*half)*MS + tj*16 + l] = c[r];
    }
  }
}

// ---------------- main kernel: one wave32 per batch element ----------------
__global__ void __launch_bounds__(WPB * 32)
ekf_kernel(const float* __restrict__ imu,        // (B,T,11,7)
           const float* __restrict__ Qc,         // (12,12)
           const float* __restrict__ prev_pose,  // (B,4,4)
           const float* __restrict__ prev_state, // (B,24)
           const float* __restrict__ prev_covar, // (B,18,18)
           const float* __restrict__ vis_meas,   // (B,T,6)
           const float* __restrict__ vmc,        // (B,T,6,6)
           const float* __restrict__ timc,       // (B,4,4)
           float* __restrict__ outP,             // (B,T+1,16)
           float* __restrict__ outS,             // (B,T+1,24)
           float* __restrict__ outC,             // (B,T+1,324)
           int Bn, int T) {
  __shared__ float smem[WPB*5*BUFSZ + BUFSZ + WPB*512];
  const int lane = threadIdx.x & 31;
  const int wid  = threadIdx.x >> 5;

  float* qbuf = smem + WPB*5*BUFSZ;            // padded 12x12 noise covar (block-shared)
  for (int i = threadIdx.x; i < BUFSZ; i += blockDim.x) qbuf[i] = 0.f;
  __syncthreads();
  for (int i = threadIdx.x; i < 144; i += blockDim.x)
    qbuf[(i/12)*MS + (i%12)] = Qc[i];
  __syncthreads();

  const int b = blockIdx.x * WPB + wid;
  if (b >= Bn) return;

  float* B0 = smem + (wid*5 + 0)*BUFSZ;
  float* B1 = smem + (wid*5 + 1)*BUFSZ;
  float* B2 = smem + (wid*5 + 2)*BUFSZ;
  float* B3 = smem + (wid*5 + 3)*BUFSZ;
  float* B4 = smem + (wid*5 + 4)*BUFSZ;        // P lives here (pads stay zero)
  float* sw   = smem + WPB*5*BUFSZ + BUFSZ + wid*512;
  float* hbuf = sw;         // 6x18
  float* hp   = sw + 108;   // 6x18
  float* kbuf = sw + 216;   // 18x6
  float* sb   = sw + 324;   // 6x6

  // ---- load carry state ----
  const float* st = prev_state + (size_t)b * 24;
  float g[3], Ck[9], rk[3], vk[3], bw[3], ba[3];
#pragma unroll
  for (int i = 0; i < 3; i++) g[i]  = st[i];
#pragma unroll
  for (int i = 0; i < 9; i++) Ck[i] = st[3 + i];
#pragma unroll
  for (int i = 0; i < 3; i++) { rk[i] = st[12+i]; vk[i] = st[15+i]; bw[i] = st[18+i]; ba[i] = st[21+i]; }

  float poseR[9], poseT[3];
  const float* pp = prev_pose + (size_t)b * 16;
#pragma unroll
  for (int r = 0; r < 3; r++) {
#pragma unroll
    for (int c = 0; c < 3; c++) poseR[r*3+c] = pp[r*4+c];
    poseT[r] = pp[r*4+3];
  }
  float Ccal[9], rcal[3], CcalT[9];
  const float* tc = timc + (size_t)b * 16;
#pragma unroll
  for (int r = 0; r < 3; r++) {
#pragma unroll
    for (int c = 0; c < 3; c++) Ccal[r*3+c] = tc[r*4+c];
    rcal[r] = tc[r*4+3];
  }
  m3T(Ccal, CcalT);

  zero32(B0, lane); zero32(B1, lane); zero32(B2, lane); zero32(B3, lane); zero32(B4, lane);
  const float* pc = prev_covar + (size_t)b * 324;
  for (int id = lane; id < 324; id += 32) B4[(id/18)*MS + id%18] = pc[id];

  // ---- emit step 0 ----
  if (lane == 0) {
    float* o = outP + ((size_t)b*(T+1) + 0)*16;
#pragma unroll
    for (int r = 0; r < 3; r++) {
#pragma unroll
      for (int c = 0; c < 3; c++) o[r*4+c] = poseR[r*3+c];
      o[r*4+3] = poseT[r];
    }
    o[12]=0.f; o[13]=0.f; o[14]=0.f; o[15]=1.f;
    float* os = outS + ((size_t)b*(T+1) + 0)*24;
    for (int i = 0; i < 24; i++) os[i] = st[i];
  }
  for (int id = lane; id < 324; id += 32)
    outC[((size_t)b*(T+1) + 0)*324 + id] = B4[(id/18)*MS + id%18];

  // ================= sequential filter =================
  for (int t = 0; t < T; ++t) {
    // -------- predict: scan over 10 IMU substeps --------
    float t_acc = 0.f;
    float Ca[9] = {1,0,0, 0,1,0, 0,0,1};
    float ra[3] = {0,0,0}, va[3] = {0,0,0};
    for (int s = 0; s < 10; ++s) {
      const float* m0 = imu + (((size_t)b*T + t)*11 + s)*7;
      const float dt  = m0[7] - m0[0];
      const float dt2 = dt * dt;
      float w[3]  = {m0[1]-bw[0], m0[2]-bw[1], m0[3]-bw[2]};
      float aa[3] = {m0[4]-ba[0], m0[5]-ba[1], m0[6]-ba[2]};
      float wsk[9];  skew3(w, wsk);
      float tm[3];
#pragma unroll
      for (int i = 0; i < 3; i++) tm[i] = vk[i] - g[i]*t_acc + va[i];
      float vv[3];   m3tvec(Ca, tm, vv);
      float vsk[9];  skew3(vv, vsk);
      float dphi[3] = {dt*w[0], dt*w[1], dt*w[2]};
      float expw[9]; expso3(dphi, expw);
      float Cvsk[9]; m3mul(Ca, vsk, Cvsk);
      float Ctg[3];  m3tvec(Ca, g, Ctg);
      float skCtg[9]; skew3(Ctg, skCtg);
      float CaT[9];  m3T(Ca, CaT);

      // F -> B0
      zero32(B0, lane);
      if (lane == 0) {
        putB3(B0, 3, 3,  wsk,  -1.f);
        putI3(B0, 3, 12, -1.f);
        putB3(B0, 6, 3,  Cvsk, -1.f);
        putB3(B0, 6, 9,  Ca,    1.f);
        putB3(B0, 9, 0,  CaT,  -1.f);
        putB3(B0, 9, 3,  skCtg,-1.f);
        putB3(B0, 9, 9,  wsk,  -1.f);
        putB3(B0, 9, 12, vsk,  -1.f);
        putI3(B0, 9, 15, -1.f);
      }
      mm32(B0, B0, B1, false, lane);                       // FF = F @ F

      // Phi -> B2
      zero32(B2, lane);
      for (int id = lane; id < 324; id += 32) {
        int i = id/18, j = id%18;
        B2[i*MS+j] = (i==j ? 1.f : 0.f) + B0[i*MS+j]*dt + 0.5f*B1[i*MS+j]*dt2;
      }
      if (lane == 0) {
        for (int r = 0; r < 3; r++)
          for (int c = 0; c < 3; c++) B2[(6+r)*MS + 12 + c] = 0.f;
        float ewT[9]; m3T(expw, ewT);
        putB3(B2, 3, 3, ewT, 1.f);
        putB3(B2, 9, 9, ewT, 1.f);
      }

      // G -> B0
      zero32(B0, lane);
      if (lane == 0) {
        putI3(B0, 3, 0, -1.f);
        putB3(B0, 9, 0, vsk, -1.f);
        putI3(B0, 9, 6, -1.f);
        putI3(B0, 12, 3, 1.f);
        putI3(B0, 15, 9, 1.f);
      }
      mm32(B2, B0,   B1, false, lane);                     // PhiG
      mm32(B1, qbuf, B3, false, lane);                     // M = PhiG @ Qc
      mm32(B3, B1,   B0, true,  lane);                     // Q = M @ PhiG^T
      mm32(B2, B4,   B3, false, lane);                     // T = Phi @ P
      mm32(B3, B2,   B1, true,  lane);                     // PP = T @ Phi^T
      for (int id = lane; id < 324; id += 32) {            // P = force_sym(PP + dt*Q)
        int i = id/18, j = id%18;
        int ii = i < j ? i : j, jj = i < j ? j : i;
        B4[i*MS+j] = B1[ii*MS+jj] + dt * B0[ii*MS+jj];
      }

      float caa[3]; m3vec(Ca, aa, caa);
#pragma unroll
      for (int i = 0; i < 3; i++) { ra[i] += va[i]*dt + 0.5f*dt2*caa[i]; va[i] += dt*caa[i]; }
      float Cn[9]; m3mul(Ca, expw, Cn);
#pragma unroll
      for (int i = 0; i < 9; i++) Ca[i] = Cn[i];
      t_acc += dt;
    }

    // predicted state
    float Cp[9]; m3mul(Ck, Ca, Cp);
    float rp[3], vt[3], vp[3];
#pragma unroll
    for (int i = 0; i < 3; i++) {
      rp[i] = rk[i] + vk[i]*t_acc - 0.5f*g[i]*t_acc*t_acc + ra[i];
      vt[i] = vk[i] - g[i]*t_acc + va[i];
    }
    m3tvec(Ca, vt, vp);

    // -------- measurement update --------
    const float* vm = vis_meas + ((size_t)b*T + t)*6;
    float vmrot[3] = {vm[0], vm[1], vm[2]};
    float vmtr[3]  = {vm[3], vm[4], vm[5]};
    float E[9];  expso3(vmrot, E);
    float A1[9]; m3mulBT(E,  Ccal, A1);    // E @ Ccal^T
    float A2[9]; m3mulBT(A1, Cp,   A2);    // @ Cp^T
    float A3[9]; m3mul(A2, Ccal, A3);      // @ Ccal
    float resrot[3]; logso3(A3, resrot);
    float t1[3]; m3vec(Cp, rcal, t1);
    float t2[3]; m3tvec(Ccal, t1, t2);
    float t3[3];
#pragma unroll
    for (int i = 0; i < 3; i++) t3[i] = rp[i] - rcal[i];
    float t4[3]; m3tvec(Ccal, t3, t4);
    float resid[6];
#pragma unroll
    for (int i = 0; i < 3; i++) { resid[i] = resrot[i]; resid[3+i] = vmtr[i] - t2[i] - t4[i]; }

    // H (sign per reference's H = -H already folded in)
    float nrr[3] = {-resrot[0], -resrot[1], -resrot[2]};
    float Jl[9];  jleftinv(nrr, Jl);
    float CtCp[9]; m3mul(CcalT, Cp, CtCp);
    float H03[9];  m3mul(Jl, CtCp, H03);          // +Jl @ Ccal^T @ Cp
    float skrc[9]; skew3(rcal, skrc);
    float H33[9];  m3mul(CtCp, skrc, H33);        // will store negated
    for (int id = lane; id < 108; id += 32) hbuf[id] = 0.f;
    if (lane == 0) {
      for (int r = 0; r < 3; r++)
        for (int c = 0; c < 3; c++) {
          hbuf[r*18 + 3 + c]     =  H03[r*3+c];
          hbuf[(3+r)*18 + 3 + c] = -H33[r*3+c];
          hbuf[(3+r)*18 + 6 + c] =  CcalT[r*3+c];
        }
    }
    // HP = H @ P
    for (int id = lane; id < 108; id += 32) {
      int i = id/18, j = id%18;
      float acc = 0.f;
      for (int k = 0; k < 18; k++) acc += hbuf[i*18+k] * B4[k*MS+j];
      hp[id] = acc;
    }
    // S = HP @ H^T + R
    const float* Rm = vmc + ((size_t)b*T + t)*36;
    for (int id = lane; id < 36; id += 32) {
      int i = id/6, j = id%6;
      float acc = Rm[id];
      for (int k = 0; k < 18; k++) acc += hp[i*18+k] * hbuf[j*18+k];
      sb[id] = acc;
    }
    // 6x6 inverse (Gauss-Jordan, redundant per lane; SPD input)
    float Msm[36], Si[36];
    for (int i = 0; i < 36; i++) { Msm[i] = sb[i]; Si[i] = (i % 7 == 0) ? 1.f : 0.f; }
    for (int col = 0; col < 6; col++) {
      float ip = 1.f / Msm[col*6+col];
      for (int j = 0; j < 6; j++) { Msm[col*6+j] *= ip; Si[col*6+j] *= ip; }
      for (int r = 0; r < 6; r++) if (r != col) {
        float f = Msm[r*6+col];
        for (int j = 0; j < 6; j++) { Msm[r*6+j] -= f*Msm[col*6+j]; Si[r*6+j] -= f*Si[col*6+j]; }
      }
    }
    // K = P H^T S^-1 = (HP)^T @ S^-1   (P symmetric)
    for (int id = lane; id < 108; id += 32) {
      int i = id/6, j = id%6;
      float acc = 0.f;
      for (int m = 0; m < 6; m++) acc += hp[m*18+i] * Si[m*6+j];
      kbuf[id] = acc;
    }
    // err = K @ residual
    float err[18];
    for (int i = 0; i < 18; i++) {
      float acc = 0.f;
      for (int j = 0; j < 6; j++) acc += kbuf[i*6+j] * resid[j];
      err[i] = acc;
    }
    // W = I - K H -> B0 ; estP = W @ P -> B1
    zero32(B0, lane);
    for (int id = lane; id < 324; id += 32) {
      int i = id/18, j = id%18;
      float acc = (i==j) ? 1.f : 0.f;
      for (int m = 0; m < 6; m++) acc -= kbuf[i*6+m] * hbuf[m*18+j];
      B0[i*MS+j] = acc;
    }
    mm32(B0, B4, B1, false, lane);

    // est state
    float ge[3];
#pragma unroll
    for (int i = 0; i < 3; i++) ge[i] = g[i] + err[i];
    float dph[3] = {err[3], err[4], err[5]};
    float Ee[9]; expso3(dph, Ee);
    float Ce[9]; m3mul(Cp, Ee, Ce);
    float re[3], ve[3], bwe[3], bae[3];
#pragma unroll
    for (int i = 0; i < 3; i++) {
      re[i]  = rp[i] + err[6+i];
      ve[i]  = vp[i] + err[9+i];
      bwe[i] = bw[i] + err[12+i];
      bae[i] = ba[i] + err[15+i];
    }

    // -------- composition --------
    float CeT[9]; m3T(Ce, CeT);
    float nR[9];  m3mul(CeT, poseR, nR);
    float pd[3];
#pragma unroll
    for (int i = 0; i < 3; i++) pd[i] = poseT[i] - re[i];
    float nT[3]; m3vec(CeT, pd, nT);
    float ng[3]; m3vec(CeT, ge, ng);
    zero32(B2, lane);
    if (lane == 0) {
      putB3(B2, 0, 0, CeT, 1.f);
      float skng[9]; skew3(ng, skng);
      putB3(B2, 0, 3, skng, 1.f);
      putI3(B2, 9, 9, 1.f); putI3(B2, 12, 12, 1.f); putI3(B2, 15, 15, 1.f);
    }
    mm32(B2, B1, B3, false, lane);   // U @ estP
    mm32(B3, B2, B0, true,  lane);   // (U estP) @ U^T
    for (int id = lane; id < 324; id += 32) {   // P = force_sym(...)
      int i = id/18, j = id%18;
      int ii = i < j ? i : j, jj = i < j ? j : i;
      B4[i*MS+j] = B0[ii*MS+jj];
    }

    // carry for next step
#pragma unroll
    for (int i = 0; i < 3; i++) {
      g[i] = ng[i]; rk[i] = 0.f; vk[i] = ve[i]; bw[i] = bwe[i]; ba[i] = bae[i]; poseT[i] = nT[i];
    }
#pragma unroll
    for (int i = 0; i < 9; i++) { Ck[i] = (i % 4 == 0) ? 1.f : 0.f; poseR[i] = nR[i]; }

    // -------- emit step t+1 --------
    if (lane == 0) {
      float* o = outP + ((size_t)b*(T+1) + t + 1)*16;
      for (int r = 0; r < 3; r++) {
        for (int c = 0; c < 3; c++) o[r*4+c] = nR[r*3+c];
        o[r*4+3] = nT[r];
      }
      o[12]=0.f; o[13]=0.f; o[14]=0.f; o[15]=1.f;
      float* os = outS + ((size_t)b*(T+1) + t + 1)*24;
      for (int i = 0; i < 3; i++) os[i] = ng[i];
      for (int i = 0; i < 9; i++) os[3+i] = (i % 4 == 0) ? 1.f : 0.f;
      for (int i = 0; i < 3; i++) os[12+i] = 0.f;
      for (int i = 0; i < 3; i++) os[15+i] = ve[i];
      for (int i = 0; i < 3; i++) os[18+i] = bwe[i];
      for (int i = 0; i < 3; i++) os[21+i] = bae[i];
    }
    for (int id = lane; id < 324; id += 32)
      outC[((size_t)b*(T+1) + t + 1)*324 + id] = B4[(id/18)*MS + id%18];
  }
}

extern "C" void kernel_launch(void* const* d_in, const int* in_sizes, int n_in,
                              void* d_out, int out_size, void* d_ws, size_t ws_size,
                              hipStream_t stream) {
  const float* imu    = (const float*)d_in[0];
  const float* qc     = (const float*)d_in[1];
  const float* ppose  = (const float*)d_in[2];
  const float* pstate = (const float*)d_in[3];
  const float* pcov   = (const float*)d_in[4];
  const float* vm     = (const float*)d_in[5];
  const float* vmcv   = (const float*)d_in[6];
  const float* timc   = (const float*)d_in[7];

  const int B = in_sizes[3] / 24;          // prev_state (B,24)
  const int T = in_sizes[5] / (6 * B);     // vis_meas (B,T,6)

  float* out  = (float*)d_out;
  float* outP = out;
  float* outS = out + (size_t)B * (T + 1) * 16;
  float* outC = outS + (size_t)B * (T + 1) * 24;

  dim3 grid((B + WPB - 1) / WPB), block(WPB * 32);
  ekf_kernel<<<grid, block, 0, stream>>>(imu, qc, ppose, pstate, pcov, vm, vmcv,
                                         timc, outP, outS, outC, B, T);
  (void)n_in; (void)out_size; (void)d_ws; (void)ws_size;
}